// SSIMLoss_87290915324392
// MI455X (gfx1250) — compile-verified
//
#include <hip/hip_runtime.h>

// ---------------------------------------------------------------------------
// SSIM loss, 16 x 1 x 1024 x 1024 fp32.
// Separable 11-tap Gaussian blurs done as banded matmuls on the CDNA5 matrix
// pipe via V_WMMA_F32_16X16X4_F32 (fp32 precision needed: sigma = E[x^2]-mu^2
// is a cancellation; 16-bit WMMA inputs would visibly corrupt it).
//
// Roofline: 128 MiB traffic -> ~5.5us at 23.3 TB/s; ~110 FMA/pixel of blur
// work would make a pure-VALU kernel compute bound, so the MACs go to WMMA.
// ---------------------------------------------------------------------------

typedef __attribute__((ext_vector_type(2))) float v2f;
typedef __attribute__((ext_vector_type(8))) float v8f;

#define W_IMG   1024
#define H_IMG   1024
#define N_IMG   16
#define TILE_W  16      // output tile width  (one wave: 16x16 WMMA tile)
#define TILE_H  128     // output tile height (8 waves stacked)
#define IN_COLS 28      // 16 + 2*5 halo, padded 26 -> 28 (7 K-chunks of 4)
#define IN_ROWS 144     // 128 + 2*5 halo, padded 138 -> 144 (9 row-tiles)
#define NKC     7       // K chunks per blur matmul

// Normalized 11-tap Gaussian, sigma = 1.5 (matches reference _gaussian_window)
__constant__ float Gw[11] = {
    0.00102838f, 0.00759875f, 0.03600077f, 0.10936068f, 0.21300553f,
    0.26601172f,
    0.21300553f, 0.10936068f, 0.03600077f, 0.00759875f, 0.00102838f
};

__device__ __forceinline__ float bandw(int d) {
    return (d >= 0 && d < 11) ? Gw[d] : 0.0f;
}

// quantity selector: 0:i1 1:i2 2:i1*i1 3:i2*i2 4:i1*i2  (q is unrolled const)
__device__ __forceinline__ float qval(int q, float a, float b) {
    switch (q) {
        case 0:  return a;
        case 1:  return b;
        case 2:  return a * a;
        case 3:  return b * b;
        default: return a * b;
    }
}

extern "C" __global__ __launch_bounds__(256)
void ssim_main(const float* __restrict__ img1,
               const float* __restrict__ img2,
               float* __restrict__ accum)
{
    __shared__ float lds1[IN_ROWS * IN_COLS];   // img1 halo tile
    __shared__ float lds2[IN_ROWS * IN_COLS];   // img2 halo tile
    __shared__ float ldsH[IN_ROWS * TILE_W];    // h-blurred slab (one quantity)
    __shared__ float ldsR[256];                 // reduction scratch

    const int tid  = threadIdx.x;
    const int lane = tid & 31;
    const int wave = tid >> 5;
    const int hi   = lane >> 4;     // lane-half selects K pair {0,1} vs {2,3}
    const int ln16 = lane & 15;     // M (A-frag) or N (B-frag) index

    const int  c0  = blockIdx.x * TILE_W;
    const int  r0  = blockIdx.y * TILE_H;
    const long nI  = (long)blockIdx.z * (long)(W_IMG * H_IMG);
    const float* p1 = img1 + nI;
    const float* p2 = img2 + nI;

    // ---- load halo tiles (zero padding outside the image, as in reference) --
    for (int idx = tid; idx < IN_ROWS * IN_COLS; idx += 256) {
        int hr  = idx / IN_COLS;
        int k   = idx - hr * IN_COLS;
        int row = r0 - 5 + hr;
        int col = c0 - 5 + k;
        bool ok = (row >= 0) && (row < H_IMG) && (col >= 0) && (col < W_IMG);
        long g  = (long)row * W_IMG + col;
        lds1[idx] = ok ? p1[g] : 0.0f;
        lds2[idx] = ok ? p2[g] : 0.0f;
    }
    __syncthreads();

    // ---- per-lane banded filter fragments --------------------------------
    // Wh[j][c] = g[j-c] (B-frag of h-pass) and Wv[m][j] = g[j-m] (A-frag of
    // v-pass) yield the identical per-lane values: band(k - (lane&15)).
    float wfx[NKC], wfy[NKC];
#pragma unroll
    for (int kc = 0; kc < NKC; ++kc) {
        int j0 = 4 * kc + 2 * hi;
        wfx[kc] = bandw(j0 - ln16);
        wfy[kc] = bandw(j0 + 1 - ln16);
    }

    v8f blur[5];

#pragma unroll
    for (int q = 0; q < 5; ++q) {
        // ---- horizontal blur: Out(16x16) = In(16x28) x Wh(28x16) ----------
        // 9 row-tiles cover the 144-row slab; wave w does tile w (+8 for w=0).
        for (int t = wave; t < 9; t += 8) {          // wave-uniform branch
            v8f acc = {};
#pragma unroll
            for (int kc = 0; kc < NKC; ++kc) {
                int k0   = 4 * kc + 2 * hi;
                int base = (16 * t + ln16) * IN_COLS + k0;
                v2f a;   // A 16x4 f32 frag: lane m = ln16, K = k0, k0+1
                a.x = qval(q, lds1[base],     lds2[base]);
                a.y = qval(q, lds1[base + 1], lds2[base + 1]);
                v2f b;   // B 4x16 frag: constant band weights
                b.x = wfx[kc];
                b.y = wfy[kc];
                acc = __builtin_amdgcn_wmma_f32_16x16x4_f32(
                          false, a, false, b, (short)0, acc, false, false);
            }
#pragma unroll
            for (int v = 0; v < 8; ++v)   // D frag: row = v + 8*hi, col = ln16
                ldsH[(16 * t + v + 8 * hi) * TILE_W + ln16] = acc[v];
        }
        __syncthreads();

        // ---- vertical blur: Out(16x16) = Wv(16x28) x H(28x16) -------------
        {
            v8f acc = {};
#pragma unroll
            for (int kc = 0; kc < NKC; ++kc) {
                int j0 = 4 * kc + 2 * hi;
                v2f a;   // A frag of Wv: band(j - m), identical to wf*
                a.x = wfx[kc];
                a.y = wfy[kc];
                v2f b;   // B frag: H rows 16*wave + j0 (+1), col ln16
                int hb = (16 * wave + j0) * TILE_W + ln16;
                b.x = ldsH[hb];
                b.y = ldsH[hb + TILE_W];
                acc = __builtin_amdgcn_wmma_f32_16x16x4_f32(
                          false, a, false, b, (short)0, acc, false, false);
            }
            blur[q] = acc;
        }
        __syncthreads();   // ldsH reused by next quantity
    }

    // ---- pointwise SSIM + block reduction --------------------------------
    const float C1v = 1.0e-4f;   // 0.01^2
    const float C2v = 9.0e-4f;   // 0.03^2
    float s = 0.0f;
#pragma unroll
    for (int v = 0; v < 8; ++v) {
        float mu1 = blur[0][v], mu2 = blur[1][v];
        float b11 = blur[2][v], b22 = blur[3][v], b12 = blur[4][v];
        float mu1sq = mu1 * mu1, mu2sq = mu2 * mu2, mu12 = mu1 * mu2;
        float s1  = b11 - mu1sq;
        float s2  = b22 - mu2sq;
        float s12 = b12 - mu12;
        float num = (2.0f * mu12 + C1v) * (2.0f * s12 + C2v);
        float den = (mu1sq + mu2sq + C1v) * (s1 + s2 + C2v);
        s += num / den;
    }
    ldsR[tid] = s;
    __syncthreads();
#pragma unroll
    for (int off = 128; off > 0; off >>= 1) {
        if (tid < off) ldsR[tid] += ldsR[tid + off];
        __syncthreads();
    }
    if (tid == 0) atomicAdd(accum, ldsR[0]);
}

extern "C" __global__ void ssim_zero(float* accum) {
    if (threadIdx.x == 0) accum[0] = 0.0f;
}

extern "C" __global__ void ssim_final(const float* __restrict__ accum,
                                      float* __restrict__ out) {
    if (threadIdx.x == 0) {
        const float invN = 1.0f / (float)((long)N_IMG * W_IMG * H_IMG);
        out[0] = 1.0f - accum[0] * invN;
    }
}

extern "C" void kernel_launch(void* const* d_in, const int* in_sizes, int n_in,
                              void* d_out, int out_size, void* d_ws, size_t ws_size,
                              hipStream_t stream) {
    const float* img1 = (const float*)d_in[0];
    const float* img2 = (const float*)d_in[1];
    float* out   = (float*)d_out;
    float* accum = (float*)d_ws;

    ssim_zero<<<1, 64, 0, stream>>>(accum);
    dim3 grid(W_IMG / TILE_W, H_IMG / TILE_H, N_IMG);   // 64 x 8 x 16
    ssim_main<<<grid, 256, 0, stream>>>(img1, img2, accum);
    ssim_final<<<1, 64, 0, stream>>>(accum, out);
}